// gnn_feature_module_62998580298149
// MI455X (gfx1250) — compile-verified
//
#include <hip/hip_runtime.h>
#include <hip/hip_bf16.h>

typedef __attribute__((ext_vector_type(2))) float v2f;
typedef __attribute__((ext_vector_type(8))) float v8f;

#define N_NODES 50000     // multiple of 16 -> 3125 row tiles, no row bounds checks
#define N_EDGES 400000
#define N_GRAPHS 64
#define C_OUT_FINAL 192

static inline int ceil_div(int a, int b) { return (a + b - 1) / b; }

// ---------------------------------------------------------------------------
// Utility kernels
// ---------------------------------------------------------------------------
__global__ void k_zero_f32(float* __restrict__ p, int n) {
    int i = blockIdx.x * blockDim.x + threadIdx.x;
    if (i < n) p[i] = 0.0f;
}

__global__ void k_count_deg(const int* __restrict__ dst, float* __restrict__ deg, int E) {
    int e = blockIdx.x * blockDim.x + threadIdx.x;
    if (e < E) atomicAdd(&deg[dst[e]], 1.0f);
}

__global__ void k_deg_inv_sqrt(float* __restrict__ dis, int n) {
    int i = blockIdx.x * blockDim.x + threadIdx.x;
    if (i < n) dis[i] = rsqrtf(dis[i] + 1.0f);
}

// Pad node features [N,3] -> [N,4] (zero 4th column) so the GEMM K loop is
// guard-free and 8-byte aligned.
__global__ void k_pad_x0(const float* __restrict__ x0, float* __restrict__ Xp, int N) {
    int i = blockIdx.x * blockDim.x + threadIdx.x;
    if (i < N) {
        const size_t s = (size_t)i * 3;
        float2* o = (float2*)(Xp + (size_t)i * 4);
        o[0] = make_float2(x0[s + 0], x0[s + 1]);
        o[1] = make_float2(x0[s + 2], 0.0f);
    }
}

// ---------------------------------------------------------------------------
// WMMA fp32 GEMM:  G = dis[row] * (X @ W), written to both H and ACC.
// One wave = one 16x16 output tile. K stepped by 4 with V_WMMA_F32_16X16X4_F32.
//  - X is [N, KPAD] row-major (pre-padded, KPAD % 4 == 0) -> A loads are b64.
//  - W staged in LDS pair-interleaved [k/2][colP][{k,k+1}] -> B loads are b64.
//  - COUT may be non-multiple-of-16 (layer 1: 24); tile over COUTP and guard
//    only the epilogue stores.
// ---------------------------------------------------------------------------
template<int KPAD, int COUT>
__launch_bounds__(256)
__global__ void k_gcn_gemm_wmma(const float* __restrict__ X,
                                const float* __restrict__ W,   // [Kdim, COUT]
                                const float* __restrict__ dis,
                                float* __restrict__ H,
                                float* __restrict__ ACC,
                                int Kdim, int tilesTotal) {
    constexpr int COUTP  = (COUT + 15) & ~15;
    constexpr int TILESN = COUTP / 16;
    __shared__ float ldsW[(KPAD / 2) * COUTP * 2];

    const int tid = threadIdx.x;
    // stage W: ldsW[(k>>1)*COUTP*2 + c*2 + (k&1)], zero-padded in k and c
    for (int idx = tid; idx < (KPAD / 2) * COUTP * 2; idx += blockDim.x) {
        const int k2  = idx / (COUTP * 2);
        const int rem = idx - k2 * (COUTP * 2);
        const int c   = rem >> 1;
        const int k   = k2 * 2 + (rem & 1);
        ldsW[idx] = (k < Kdim && c < COUT) ? W[k * COUT + c] : 0.0f;
    }
    __syncthreads();

    const int wave = tid >> 5;
    const int lane = tid & 31;
    const int tile = blockIdx.x * (blockDim.x >> 5) + wave;
    if (tile >= tilesTotal) return;          // wave-uniform branch, EXEC stays all-1s

    const int tn = tile % TILESN;
    const int tm = tile / TILESN;

    const int halfSel = lane >> 4;           // 0: lanes 0-15 (K 0,1) / 1: lanes 16-31 (K 2,3)
    const int l15     = lane & 15;
    const int row     = tm * 16 + l15;       // always < N (N multiple of 16)
    const int col     = tn * 16 + l15;

    const float2* __restrict__ Arow = (const float2*)(X + (size_t)row * KPAD);
    const float2* __restrict__ Blds = (const float2*)ldsW;

    v8f acc = {};
#pragma unroll
    for (int kk = 0; kk < KPAD; kk += 4) {
        const int p = (kk >> 1) + halfSel;   // k-pair index for this lane
        const float2 af = Arow[p];           // global_load_b64
        const float2 bf = Blds[p * COUTP + col];  // ds_load_b64
        v2f a; a.x = af.x; a.y = af.y;
        v2f b; b.x = bf.x; b.y = bf.y;
        acc = __builtin_amdgcn_wmma_f32_16x16x4_f32(
            /*neg_a=*/false, a, /*neg_b=*/false, b,
            /*c_mod=*/(short)0, acc, /*reuse_a=*/false, /*reuse_b=*/false);
    }

    if (col < COUT) {                        // only ragged for COUT==24
#pragma unroll
        for (int i = 0; i < 8; ++i) {
            const int r = tm * 16 + i + halfSel * 8;   // C/D layout: vgpr i, halves M=i / M=i+8
            const float g = dis[r] * acc[i];
            const size_t o = (size_t)r * COUT + col;
            H[o]   = g;   // message value for edges with src == r
            ACC[o] = g;   // self-loop term (init of aggregation)
        }
    }
}

// ---------------------------------------------------------------------------
// Edge scatter-add: ACC[dst, c] += H[src, c]
// ---------------------------------------------------------------------------
template<int C>
__global__ void k_gcn_scatter(const float* __restrict__ H,
                              float* __restrict__ ACC,
                              const int* __restrict__ src,
                              const int* __restrict__ dst,
                              int E) {
    long long idx = (long long)blockIdx.x * blockDim.x + threadIdx.x;
    if (idx >= (long long)E * C) return;
    const int e = (int)(idx / C);
    const int c = (int)(idx - (long long)e * C);
    atomicAdd(&ACC[(size_t)dst[e] * C + c], H[(size_t)src[e] * C + c]);
}

// ---------------------------------------------------------------------------
// Finalize: Xn[i,c] = dis[i] * ACC[i,c] + b[c]
// ---------------------------------------------------------------------------
template<int C>
__global__ void k_gcn_finalize(const float* __restrict__ ACC,
                               const float* __restrict__ dis,
                               const float* __restrict__ b,
                               float* __restrict__ Xn, int N) {
    long long idx = (long long)blockIdx.x * blockDim.x + threadIdx.x;
    if (idx >= (long long)N * C) return;
    const int i = (int)(idx / C);
    const int c = (int)(idx - (long long)i * C);
    Xn[idx] = dis[i] * ACC[idx] + b[c];
}

// ---------------------------------------------------------------------------
// Pooling
// ---------------------------------------------------------------------------
template<int C>
__global__ void k_pool_sum(const float* __restrict__ X,
                           const int* __restrict__ batch,
                           float* __restrict__ out, int N) {
    long long idx = (long long)blockIdx.x * blockDim.x + threadIdx.x;
    if (idx >= (long long)N * C) return;
    const int i = (int)(idx / C);
    const int c = (int)(idx - (long long)i * C);
    atomicAdd(&out[(size_t)batch[i] * C + c], X[idx]);
}

__global__ void k_pool_count(const int* __restrict__ batch,
                             float* __restrict__ counts, int N) {
    int i = blockIdx.x * blockDim.x + threadIdx.x;
    if (i < N) atomicAdd(&counts[batch[i]], 1.0f);
}

__global__ void k_pool_div(float* __restrict__ out,
                           const float* __restrict__ counts, int G, int C) {
    int idx = blockIdx.x * blockDim.x + threadIdx.x;
    if (idx >= G * C) return;
    out[idx] = out[idx] / fmaxf(counts[idx / C], 1.0f);
}

// ---------------------------------------------------------------------------
// Launch
// ---------------------------------------------------------------------------
extern "C" void kernel_launch(void* const* d_in, const int* in_sizes, int n_in,
                              void* d_out, int out_size, void* d_ws, size_t ws_size,
                              hipStream_t stream) {
    const float* x0    = (const float*)d_in[0];                 // [N, 3]
    const int*   edges = (const int*)d_in[1];                   // [2, E]
    const int*   batch = (const int*)d_in[2];                   // [N]
    const float* W1 = (const float*)d_in[3]; const float* b1 = (const float*)d_in[4];
    const float* W2 = (const float*)d_in[5]; const float* b2 = (const float*)d_in[6];
    const float* W3 = (const float*)d_in[7]; const float* b3 = (const float*)d_in[8];

    const int* srcIdx = edges;              // row 0
    const int* dstIdx = edges + N_EDGES;    // row 1

    // workspace layout (floats)
    const size_t featCap = (size_t)N_NODES * C_OUT_FINAL;       // 9.6 M floats
    float* Xbuf = (float*)d_ws;                                 // layer inputs / padded x0
    float* Hbuf = Xbuf + featCap;
    float* Abuf = Hbuf + featCap;
    float* dis  = Abuf + featCap;                               // N_NODES (also deg)
    float* cnts = dis + N_NODES;                                // N_GRAPHS
    float* outf = (float*)d_out;                                // [64, 192]

    const int B = 256;
    const int WPB = B / 32;                                     // waves per block

    // ---- degrees -> dis ----
    k_zero_f32<<<ceil_div(N_NODES, B), B, 0, stream>>>(dis, N_NODES);
    k_count_deg<<<ceil_div(N_EDGES, B), B, 0, stream>>>(dstIdx, dis, N_EDGES);
    k_deg_inv_sqrt<<<ceil_div(N_NODES, B), B, 0, stream>>>(dis, N_NODES);

    const int rowTiles = N_NODES / 16;                          // 3125 exact

    // ---- layer 1: K=3 (padded to 4), Cout=24 ----
    k_pad_x0<<<ceil_div(N_NODES, B), B, 0, stream>>>(x0, Xbuf, N_NODES);
    {
        const int tiles = rowTiles * 2;                         // COUTP=32 -> 2 col tiles
        k_gcn_gemm_wmma<4, 24><<<ceil_div(tiles, WPB), B, 0, stream>>>(
            Xbuf, W1, dis, Hbuf, Abuf, 3, tiles);
        k_gcn_scatter<24><<<(int)(((long long)N_EDGES * 24 + B - 1) / B), B, 0, stream>>>(
            Hbuf, Abuf, srcIdx, dstIdx, N_EDGES);
        k_gcn_finalize<24><<<(int)(((long long)N_NODES * 24 + B - 1) / B), B, 0, stream>>>(
            Abuf, dis, b1, Xbuf, N_NODES);
    }

    // ---- layer 2: K=24, Cout=48 ----
    {
        const int tiles = rowTiles * 3;
        k_gcn_gemm_wmma<24, 48><<<ceil_div(tiles, WPB), B, 0, stream>>>(
            Xbuf, W2, dis, Hbuf, Abuf, 24, tiles);
        k_gcn_scatter<48><<<(int)(((long long)N_EDGES * 48 + B - 1) / B), B, 0, stream>>>(
            Hbuf, Abuf, srcIdx, dstIdx, N_EDGES);
        k_gcn_finalize<48><<<(int)(((long long)N_NODES * 48 + B - 1) / B), B, 0, stream>>>(
            Abuf, dis, b2, Xbuf, N_NODES);
    }

    // ---- layer 3: K=48, Cout=192 ----
    {
        const int tiles = rowTiles * 12;
        k_gcn_gemm_wmma<48, 192><<<ceil_div(tiles, WPB), B, 0, stream>>>(
            Xbuf, W3, dis, Hbuf, Abuf, 48, tiles);
        k_gcn_scatter<192><<<(int)(((long long)N_EDGES * 192 + B - 1) / B), B, 0, stream>>>(
            Hbuf, Abuf, srcIdx, dstIdx, N_EDGES);
        k_gcn_finalize<192><<<(int)(((long long)N_NODES * 192 + B - 1) / B), B, 0, stream>>>(
            Abuf, dis, b3, Xbuf, N_NODES);
    }

    // ---- mean pooling per graph ----
    k_zero_f32<<<ceil_div(N_GRAPHS * C_OUT_FINAL, B), B, 0, stream>>>(outf, N_GRAPHS * C_OUT_FINAL);
    k_zero_f32<<<1, B, 0, stream>>>(cnts, N_GRAPHS);
    k_pool_sum<192><<<(int)(((long long)N_NODES * 192 + B - 1) / B), B, 0, stream>>>(
        Xbuf, batch, outf, N_NODES);
    k_pool_count<<<ceil_div(N_NODES, B), B, 0, stream>>>(batch, cnts, N_NODES);
    k_pool_div<<<ceil_div(N_GRAPHS * C_OUT_FINAL, B), B, 0, stream>>>(outf, cnts, N_GRAPHS, C_OUT_FINAL);
}